// MoM_87574383166010
// MI455X (gfx1250) — compile-verified
//
#include <hip/hip_runtime.h>
#include <hip/hip_bf16.h>

typedef __attribute__((ext_vector_type(2))) float v2f;
typedef __attribute__((ext_vector_type(8))) float v8f;
typedef __attribute__((ext_vector_type(4))) unsigned int v4u;
typedef __attribute__((ext_vector_type(8))) int v8i;
typedef __attribute__((ext_vector_type(4))) int v4i;

#define T_DIM 2048
#define B_DIM 8
#define D_DIM 1024
#define HD    64
#define NMEM  8
#define MP1   9
#define NTOK  (T_DIM * B_DIM)   /* 16384 */
#define TT    (NTOK * 3)        /* 49152 packed tokens */
#define NSEG  (B_DIM * MP1)     /* 72 segments */

// ---------------- workspace layout (bytes) ----------------
constexpr size_t O_IDX  = 0;                               // int  [NTOK*2]
constexpr size_t O_ALPH = O_IDX  + (size_t)NTOK * 2 * 4;   // f32  [NTOK*2]
constexpr size_t O_CNT  = O_ALPH + (size_t)NTOK * 2 * 4;   // int  [72]
constexpr size_t O_OFF  = O_CNT  + 512;                    // int  [73]
constexpr size_t O_RPK  = O_OFF  + 512;                    // int  [TT]
constexpr size_t O_APK  = O_RPK  + (size_t)TT * 4;         // f32  [TT]
constexpr size_t O_QPK  = O_APK  + (size_t)TT * 4;         // f32  [TT*64]
constexpr size_t O_KPK  = O_QPK  + (size_t)TT * HD * 4;    // f32  [TT*64]
constexpr size_t O_VPK  = O_KPK  + (size_t)TT * HD * 4;    // f32  [TT*64]
constexpr size_t O_OPK  = O_VPK  + (size_t)TT * HD * 4;    // f32  [TT*64]
constexpr size_t O_INV  = O_OPK  + (size_t)TT * HD * 4;    // int  [NTOK*3]

// exact-f32 CDNA5 WMMA: D(16x16) = A(16x4) * B(4x16) + C
static __device__ __forceinline__ v8f wmma_f32(v2f a, v2f b, v8f c) {
  return __builtin_amdgcn_wmma_f32_16x16x4_f32(false, a, false, b, (short)0, c, false, false);
}

// ---- Tensor Data Mover: 2D tile (f32 elements) global -> LDS, per ISA cdna5 ch.8 D# ----
// pad_en: insert pad_amount+1 dwords into LDS dest after every (1<<pad_iv) 8-byte units.
// Rows beyond tens_h are hardware zero-filled (OOB reads return zero).
static __device__ __forceinline__ void tdm_load_2d(
    const float* gptr, unsigned int lds_byte, unsigned int tile_w,
    unsigned int tile_h, unsigned int tens_w, unsigned int tens_h,
    unsigned int stride_elts, unsigned int pad_iv, unsigned int pad_amt,
    unsigned int pad_en)
{
  const unsigned long long ga = (unsigned long long)(uintptr_t)gptr;
  v4u g0;
  g0[0] = 1u;                                              // count=1, user descriptor
  g0[1] = lds_byte;                                        // lds_addr
  g0[2] = (unsigned int)ga;                                // global_addr[31:0]
  g0[3] = (unsigned int)((ga >> 32) & 0x1FFFFFFu) | (2u << 30); // addr[56:32] | type=2
  unsigned int d0 = (2u << 16);                            // data_size = 4 bytes
  if (pad_en) d0 |= (1u << 20) | (pad_iv << 22) | (pad_amt << 25);
  v8i g1;
  g1[0] = (int)d0;
  g1[1] = (int)((tens_w & 0xFFFFu) << 16);                 // tensor_dim0[15:0] (bits 79:64..)
  g1[2] = (int)((tens_w >> 16) | ((tens_h & 0xFFFFu) << 16));
  g1[3] = (int)((tens_h >> 16) | (tile_w << 16));          // tile_dim0
  g1[4] = (int)(tile_h & 0xFFFFu);                         // tile_dim1 (tile_dim2=0 -> 2D)
  g1[5] = (int)stride_elts;                                // tensor_dim0_stride lo32
  g1[6] = 0;
  g1[7] = 0;
  const v4i gz4 = {0, 0, 0, 0};
  const v8i gz8 = {0, 0, 0, 0, 0, 0, 0, 0};
  // clang-23 / amdgpu-toolchain form: 6 args (extra int32x8 group before cpol)
  __builtin_amdgcn_tensor_load_to_lds(g0, g1, gz4, gz4, gz8, 0);
}

static __device__ __forceinline__ unsigned int lds_off(const void* p) {
  return (unsigned int)(uintptr_t)p;   // LDS aperture: addr[31:0] is the LDS byte offset
}

// ---------------- 1) gating: softmax + top-2 ----------------
__global__ __launch_bounds__(256) void gate_kernel(
    const float* __restrict__ X, const float* __restrict__ Wg,
    const float* __restrict__ bg, int* __restrict__ idx_top,
    float* __restrict__ alpha_top)
{
  __shared__ float wg_s[D_DIM * NMEM]; // 32KB
  const int tid = threadIdx.x;
  for (int i = tid; i < D_DIM * NMEM; i += 256) wg_s[i] = Wg[i];
  __syncthreads();

  const int wv = tid >> 5, lane = tid & 31;
  const int tok = blockIdx.x * 8 + wv;       // one wave per token
  const float* xr = X + (size_t)tok * D_DIM;

  float acc[NMEM];
#pragma unroll
  for (int j = 0; j < NMEM; ++j) acc[j] = 0.f;

  for (int i = 0; i < D_DIM / 32; ++i) {
    const int c = lane + 32 * i;
    const float xv = xr[c];
#pragma unroll
    for (int j = 0; j < NMEM; ++j) acc[j] += xv * wg_s[c * NMEM + j];
  }
#pragma unroll
  for (int j = 0; j < NMEM; ++j)
    for (int off = 16; off; off >>= 1) acc[j] += __shfl_xor(acc[j], off, 32);

  if (lane == 0) {
    float l[NMEM];
#pragma unroll
    for (int j = 0; j < NMEM; ++j) l[j] = acc[j] + bg[j];
    float mx = l[0];
#pragma unroll
    for (int j = 1; j < NMEM; ++j) mx = fmaxf(mx, l[j]);
    float e[NMEM];
#pragma unroll
    for (int j = 0; j < NMEM; ++j) e[j] = expf(l[j] - mx);
    int i1 = 0; float v1 = e[0];
#pragma unroll
    for (int j = 1; j < NMEM; ++j) if (e[j] > v1) { v1 = e[j]; i1 = j; }
    int i2 = -1; float v2 = -1.f;
#pragma unroll
    for (int j = 0; j < NMEM; ++j) if (j != i1 && e[j] > v2) { v2 = e[j]; i2 = j; }
    const float denom = v1 + v2;
    idx_top[tok * 2 + 0] = i1 + 1;
    idx_top[tok * 2 + 1] = i2 + 1;
    alpha_top[tok * 2 + 0] = v1 / denom;
    alpha_top[tok * 2 + 1] = v2 / denom;
  }
}

// ---------------- 2) segment histogram ----------------
__global__ void count_kernel(const int* __restrict__ idx_top, int* __restrict__ cnt)
{
  const int tok = blockIdx.x * 256 + threadIdx.x;
  if (tok >= NTOK) return;
  const int b = tok & (B_DIM - 1);
  atomicAdd(&cnt[b * MP1 + 0], 1);
  atomicAdd(&cnt[b * MP1 + idx_top[tok * 2 + 0]], 1);
  atomicAdd(&cnt[b * MP1 + idx_top[tok * 2 + 1]], 1);
}

// ---------------- 3) exclusive scan over 72 segments ----------------
__global__ void scan_kernel(const int* __restrict__ cnt, int* __restrict__ off)
{
  if (threadIdx.x == 0) {
    int acc = 0;
    for (int i = 0; i < NSEG; ++i) { off[i] = acc; acc += cnt[i]; }
    off[NSEG] = acc;
  }
}

// ---------------- 4) stable pack (time order inside each segment) ----------------
__global__ __launch_bounds__(256) void pack_kernel(
    const int* __restrict__ idx_top, const float* __restrict__ alpha_top,
    const int* __restrict__ seg_off, int* __restrict__ r_pk,
    float* __restrict__ alpha_pk, int* __restrict__ inv)
{
  __shared__ int wsum[8];
  __shared__ int running;
  const int s = blockIdx.x;        // segment = b*9 + m
  const int b = s / MP1, m = s % MP1;
  const int base = seg_off[s];
  const int tid = threadIdx.x, lane = tid & 31, w = tid >> 5;
  if (tid == 0) running = 0;
  __syncthreads();

  for (int t0 = 0; t0 < T_DIM; t0 += 256) {
    const int t = t0 + tid;
    const int tok = t * B_DIM + b;
    int f = 0, slot = 0; float al = 0.f;
    if (m == 0) { f = 1; al = 1.f; slot = 0; }
    else {
      const int i2 = tok * 2;
      if (idx_top[i2] == m)          { f = 1; al = alpha_top[i2];     slot = 1; }
      else if (idx_top[i2 + 1] == m) { f = 1; al = alpha_top[i2 + 1]; slot = 2; }
    }
    const unsigned bm = (unsigned)__ballot(f);
    const int pre = __popc(bm & ((1u << lane) - 1u));
    if (lane == 0) wsum[w] = __popc(bm);
    __syncthreads();
    int woff = 0;
#pragma unroll
    for (int i = 0; i < 8; ++i) if (i < w) woff += wsum[i];
    int tot = 0;
#pragma unroll
    for (int i = 0; i < 8; ++i) tot += wsum[i];
    if (f) {
      const int p = base + running + woff + pre;
      r_pk[p] = tok;
      alpha_pk[p] = al;
      inv[tok * 3 + slot] = p;
    }
    __syncthreads();
    if (tid == 0) running += tot;
    __syncthreads();
  }
}

// ---------------- 5) gathered projection GEMM (q / routed-head k,v), f32 WMMA ----------------
// A tile: per-lane gathered loads (indexed rows). W tiles: TDM tensor_load_to_lds,
// natural [k][n] layout (B-frag row pairs -> ds_load_2addr, conflict-free).
__global__ __launch_bounds__(128) void proj_kernel(
    const float* __restrict__ X,
    const float* __restrict__ Wq, const float* __restrict__ bq,
    const float* __restrict__ Wk, const float* __restrict__ bk,
    const float* __restrict__ Wv, const float* __restrict__ bv,
    const int* __restrict__ seg_off, const int* __restrict__ r_pk,
    float* __restrict__ q_pk, float* __restrict__ k_pk, float* __restrict__ v_pk)
{
  __shared__ float As[64 * 33];   // gathered X tile, [row][k], padded
  __shared__ float Bqs[32 * 64];  // weight tiles, natural [k][n]
  __shared__ float Bks[32 * 64];
  __shared__ float Bvs[32 * 64];
  __shared__ int rr[64];

  const int s = blockIdx.y;
  const int m = s % MP1;
  const int p0 = seg_off[s] + blockIdx.x * 64;
  const int pend = seg_off[s + 1];
  if (p0 >= pend) return;

  const int tid = threadIdx.x;
  const int lane = tid & 31, w = tid >> 5, wrow = w * 16;
  const int mr = lane & 15, kh = lane >> 4;

  const unsigned int bq_lds = lds_off(Bqs);
  const unsigned int bk_lds = lds_off(Bks);
  const unsigned int bv_lds = lds_off(Bvs);

  if (tid < 64) {
    const int p = p0 + tid;
    rr[tid] = (p < pend) ? r_pk[p] : r_pk[p0];
  }
  __syncthreads();

  const v8f vzero = {0.f, 0.f, 0.f, 0.f, 0.f, 0.f, 0.f, 0.f};
  v8f cq[4], ck[4], cv[4];
#pragma unroll
  for (int nt = 0; nt < 4; ++nt) { cq[nt] = vzero; ck[nt] = vzero; cv[nt] = vzero; }

  for (int kc = 0; kc < D_DIM; kc += 32) {
    // kick off TDM for the three 32x64 weight tiles of this k-slab
    if (w == 0) {
      tdm_load_2d(Wq + (size_t)kc * HD,                 bq_lds, 64, 32, 64, 32, HD,  0, 0, 0);
      tdm_load_2d(Wk + (size_t)kc * (MP1*HD) + m * HD,  bk_lds, 64, 32, 64, 32, MP1*HD, 0, 0, 0);
      tdm_load_2d(Wv + (size_t)kc * (MP1*HD) + m * HD,  bv_lds, 64, 32, 64, 32, MP1*HD, 0, 0, 0);
    }
    // overlap: all waves stage the gathered A tile (64 rows x 32 k)
#pragma unroll
    for (int it = 0; it < 4; ++it) {
      const int idx = tid + 128 * it;
      const int row = idx >> 3, q4 = idx & 7;
      const float* xp = X + (size_t)rr[row] * D_DIM + kc + q4 * 4;
      const float4 xv = *(const float4*)xp;
      if (kc + 32 < D_DIM) __builtin_prefetch(xp + 32, 0, 1);
      float* a = &As[row * 33 + q4 * 4];
      a[0] = xv.x; a[1] = xv.y; a[2] = xv.z; a[3] = xv.w;
    }
    if (w == 0) __builtin_amdgcn_s_wait_tensorcnt((short)0);
    __syncthreads();
#pragma unroll
    for (int kk = 0; kk < 32; kk += 4) {
      const int ak = kk + 2 * kh;
      v2f a; a.x = As[(wrow + mr) * 33 + ak]; a.y = As[(wrow + mr) * 33 + ak + 1];
#pragma unroll
      for (int nt = 0; nt < 4; ++nt) {
        const int cb = ak * 64 + nt * 16 + mr;
        v2f b0; b0.x = Bqs[cb]; b0.y = Bqs[cb + 64];
        cq[nt] = wmma_f32(a, b0, cq[nt]);
        v2f b1; b1.x = Bks[cb]; b1.y = Bks[cb + 64];
        ck[nt] = wmma_f32(a, b1, ck[nt]);
        v2f b2; b2.x = Bvs[cb]; b2.y = Bvs[cb + 64];
        cv[nt] = wmma_f32(a, b2, cv[nt]);
      }
    }
    __syncthreads();
  }
  // epilogue: + bias, store packed
#pragma unroll
  for (int nt = 0; nt < 4; ++nt) {
    const int col = nt * 16 + mr;
    const float bqv = bq[col];
    const float bkv = bk[m * HD + col];
    const float bvv = bv[m * HD + col];
#pragma unroll
    for (int i = 0; i < 8; ++i) {
      const int row = wrow + i + 8 * kh;
      const int p = p0 + row;
      if (p < pend) {
        q_pk[(size_t)p * HD + col] = cq[nt][i] + bqv;
        k_pk[(size_t)p * HD + col] = ck[nt][i] + bkv;
        v_pk[(size_t)p * HD + col] = cv[nt][i] + bvv;
      }
    }
  }
}

// ---------------- 6) chunked per-segment linear attention, f32 WMMA ----------------
// per chunk (C=64): O = Q @ M_prev + tril(Q K^T) @ V ; M_prev += K^T V
// Q/K/V chunk tiles arrive via double-buffered TDM loads (pad_enable gives the
// stride-65 anti-conflict padding; tensor_dim1 = nv gives hardware zero-fill).
__global__ __launch_bounds__(128) void seg_attn_kernel(
    const float* __restrict__ M0,
    const float* __restrict__ q_pk, const float* __restrict__ k_pk,
    const float* __restrict__ v_pk, const float* __restrict__ alpha_pk,
    const int* __restrict__ seg_off, float* __restrict__ o_pk)
{
  __shared__ float Ms[64 * 64];        // running state
  __shared__ float Qs[2][64 * 65];     // Q (later overwritten row-block-wise by S)
  __shared__ float Ks[2][64 * 65];
  __shared__ float Vs[2][64 * 64];
  __shared__ float alpha_l[64];

  const int s = blockIdx.x;
  const int p0s = seg_off[s], pend = seg_off[s + 1];
  if (p0s >= pend) return;

  const int tid = threadIdx.x;
  const int lane = tid & 31, w = tid >> 5, wrow = w * 16;
  const int mr = lane & 15, kh = lane >> 4;
  const v8f vzero = {0.f, 0.f, 0.f, 0.f, 0.f, 0.f, 0.f, 0.f};

  // prologue: TDM-load the first chunk while other waves stage M0
  int cur = 0;
  if (w == 0) {
    const unsigned int nv0 = (unsigned int)min(64, pend - p0s);
    tdm_load_2d(q_pk + (size_t)p0s * HD, lds_off(Qs[0]), 64, 64, 64, nv0, HD, 5, 0, 1);
    tdm_load_2d(k_pk + (size_t)p0s * HD, lds_off(Ks[0]), 64, 64, 64, nv0, HD, 5, 0, 1);
    tdm_load_2d(v_pk + (size_t)p0s * HD, lds_off(Vs[0]), 64, 64, 64, nv0, HD, 0, 0, 0);
  }
  for (int i = tid; i < 64 * 64; i += 128) Ms[i] = M0[i];
  if (w == 0) __builtin_amdgcn_s_wait_tensorcnt((short)0);
  __syncthreads();

  for (int p0 = p0s; p0 < pend; p0 += 64) {
    const int nv = min(64, pend - p0);
    const bool has_next = (p0 + 64) < pend;
    // prefetch next chunk into the other buffer via TDM
    if (w == 0 && has_next) {
      const int pn = p0 + 64;
      const unsigned int nvn = (unsigned int)min(64, pend - pn);
      tdm_load_2d(q_pk + (size_t)pn * HD, lds_off(Qs[cur ^ 1]), 64, 64, 64, nvn, HD, 5, 0, 1);
      tdm_load_2d(k_pk + (size_t)pn * HD, lds_off(Ks[cur ^ 1]), 64, 64, 64, nvn, HD, 5, 0, 1);
      tdm_load_2d(v_pk + (size_t)pn * HD, lds_off(Vs[cur ^ 1]), 64, 64, 64, nvn, HD, 0, 0, 0);
    }
    if (tid < 64) {
      const int p = p0 + tid;
      alpha_l[tid] = (p < pend) ? alpha_pk[p] : 0.f;
    }
    __syncthreads();

    float* Qb = Qs[cur];
    float* Kb = Ks[cur];
    float* Vb = Vs[cur];

    // phase 1: S = tril(Q K^T) for this wave's 16 rows
    v8f sf[4];
#pragma unroll
    for (int nt = 0; nt < 4; ++nt) {
      sf[nt] = vzero;
      if (nt > w) continue;                 // strictly above diagonal block
#pragma unroll
      for (int kk = 0; kk < 64; kk += 4) {
        const int ak = kk + 2 * kh;
        v2f a;  a.x = Qb[(wrow + mr) * 65 + ak];  a.y = Qb[(wrow + mr) * 65 + ak + 1];
        v2f bb; bb.x = Kb[(nt * 16 + mr) * 65 + ak]; bb.y = Kb[(nt * 16 + mr) * 65 + ak + 1];
        sf[nt] = wmma_f32(a, bb, sf[nt]);
      }
      if (nt == w) {                        // diagonal block: causal mask (inclusive)
#pragma unroll
        for (int i = 0; i < 8; ++i) {
          const int rg = wrow + i + 8 * kh, cg = nt * 16 + mr;
          if (cg > rg) sf[nt][i] = 0.f;
        }
      }
    }
    // phase 2a: O = Q @ M_prev
    v8f of[4];
#pragma unroll
    for (int nt = 0; nt < 4; ++nt) {
      of[nt] = vzero;
#pragma unroll
      for (int kk = 0; kk < 64; kk += 4) {
        const int ak = kk + 2 * kh;
        v2f a;  a.x = Qb[(wrow + mr) * 65 + ak];  a.y = Qb[(wrow + mr) * 65 + ak + 1];
        v2f bb; bb.x = Ms[ak * 64 + nt * 16 + mr]; bb.y = Ms[(ak + 1) * 64 + nt * 16 + mr];
        of[nt] = wmma_f32(a, bb, of[nt]);
      }
    }
    // overwrite this wave's own Q rows with S (rows are wave-private)
#pragma unroll
    for (int nt = 0; nt < 4; ++nt)
#pragma unroll
      for (int i = 0; i < 8; ++i)
        Qb[(wrow + i + 8 * kh) * 65 + nt * 16 + mr] = sf[nt][i];
    // phase 2b: O += S @ V
#pragma unroll
    for (int nt = 0; nt < 4; ++nt) {
#pragma unroll
      for (int kk = 0; kk < 64; kk += 4) {
        const int ak = kk + 2 * kh;
        v2f a;  a.x = Qb[(wrow + mr) * 65 + ak];  a.y = Qb[(wrow + mr) * 65 + ak + 1];
        v2f bb; bb.x = Vb[ak * 64 + nt * 16 + mr]; bb.y = Vb[(ak + 1) * 64 + nt * 16 + mr];
        of[nt] = wmma_f32(a, bb, of[nt]);
      }
    }
    // write alpha-scaled per-packed-position outputs (no atomics -> deterministic)
#pragma unroll
    for (int nt = 0; nt < 4; ++nt) {
      const int cg = nt * 16 + mr;
#pragma unroll
      for (int i = 0; i < 8; ++i) {
        const int row = wrow + i + 8 * kh;
        if (row < nv)
          o_pk[(size_t)(p0 + row) * HD + cg] = alpha_l[row] * of[nt][i];
      }
    }
    __syncthreads();
    // phase 3: M_prev += K^T V (this wave owns state rows wrow..wrow+15)
#pragma unroll
    for (int nt = 0; nt < 4; ++nt) {
      v8f dm = vzero;
#pragma unroll
      for (int kk = 0; kk < 64; kk += 4) {
        const int ak = kk + 2 * kh;
        v2f a;  a.x = Kb[ak * 65 + wrow + mr];       a.y = Kb[(ak + 1) * 65 + wrow + mr];
        v2f bb; bb.x = Vb[ak * 64 + nt * 16 + mr];   bb.y = Vb[(ak + 1) * 64 + nt * 16 + mr];
        dm = wmma_f32(a, bb, dm);
      }
#pragma unroll
      for (int i = 0; i < 8; ++i)
        Ms[(wrow + i + 8 * kh) * 64 + nt * 16 + mr] += dm[i];
    }
    if (w == 0 && has_next) __builtin_amdgcn_s_wait_tensorcnt((short)0);
    __syncthreads();
    cur ^= 1;
  }
}

// ---------------- 7) combine: each token sums its exactly-3 contributions ----------------
__global__ void combine_kernel(const float* __restrict__ o_pk,
                               const int* __restrict__ inv,
                               float* __restrict__ out)
{
  const int idx = blockIdx.x * 256 + threadIdx.x;
  if (idx >= NTOK * HD) return;
  const int tok = idx >> 6, j = idx & 63;
  float r = 0.f;
#pragma unroll
  for (int sl = 0; sl < 3; ++sl)
    r += o_pk[(size_t)inv[tok * 3 + sl] * HD + j];
  out[idx] = r;
}

// ---------------- launch ----------------
extern "C" void kernel_launch(void* const* d_in, const int* in_sizes, int n_in,
                              void* d_out, int out_size, void* d_ws, size_t ws_size,
                              hipStream_t stream) {
  (void)in_sizes; (void)n_in; (void)out_size; (void)ws_size;
  const float* X  = (const float*)d_in[0];
  const float* M0 = (const float*)d_in[1];
  const float* Wq = (const float*)d_in[2];
  const float* bq = (const float*)d_in[3];
  const float* Wk = (const float*)d_in[4];
  const float* bk = (const float*)d_in[5];
  const float* Wv = (const float*)d_in[6];
  const float* bv = (const float*)d_in[7];
  const float* Wg = (const float*)d_in[8];
  const float* bg = (const float*)d_in[9];
  float* out = (float*)d_out;
  char* ws = (char*)d_ws;

  int*   idx_top  = (int*)  (ws + O_IDX);
  float* alpha_tp = (float*)(ws + O_ALPH);
  int*   seg_cnt  = (int*)  (ws + O_CNT);
  int*   seg_off  = (int*)  (ws + O_OFF);
  int*   r_pk     = (int*)  (ws + O_RPK);
  float* alpha_pk = (float*)(ws + O_APK);
  float* q_pk     = (float*)(ws + O_QPK);
  float* k_pk     = (float*)(ws + O_KPK);
  float* v_pk     = (float*)(ws + O_VPK);
  float* o_pk     = (float*)(ws + O_OPK);
  int*   inv      = (int*)  (ws + O_INV);

  (void)hipMemsetAsync(seg_cnt, 0, 512, stream);
  gate_kernel<<<NTOK / 8, 256, 0, stream>>>(X, Wg, bg, idx_top, alpha_tp);
  count_kernel<<<NTOK / 256, 256, 0, stream>>>(idx_top, seg_cnt);
  scan_kernel<<<1, 32, 0, stream>>>(seg_cnt, seg_off);
  pack_kernel<<<NSEG, 256, 0, stream>>>(idx_top, alpha_tp, seg_off, r_pk, alpha_pk, inv);
  proj_kernel<<<dim3(32, NSEG), 128, 0, stream>>>(X, Wq, bq, Wk, bk, Wv, bv,
                                                  seg_off, r_pk, q_pk, k_pk, v_pk);
  seg_attn_kernel<<<NSEG, 128, 0, stream>>>(M0, q_pk, k_pk, v_pk, alpha_pk, seg_off, o_pk);
  combine_kernel<<<(NTOK * HD + 255) / 256, 256, 0, stream>>>(o_pk, inv, out);
}